// JointEdgeSegLoss_21294447853991
// MI455X (gfx1250) — compile-verified
//
#include <hip/hip_runtime.h>

#define NCLASSES 19
#define NIMG 4
#define HH 512
#define WW 1024
#define HW (HH * WW)        // 524288 = 2^19
#define PIX (NIMG * HW)     // 2097152
#define QPIX (PIX / 4)      // 524288 quad-pixels
#define QHW (HW / 4)        // 131072 = 2^17
#define IGNORE_LBL 255

typedef __attribute__((ext_vector_type(2))) float v2f;
typedef __attribute__((ext_vector_type(8))) float v8f;

// ---- workspace layout -------------------------------------------------
struct Ws {
  unsigned int seg_bins[NCLASSES];
  unsigned int att_bins[NCLASSES];
  unsigned int pos_num;
  unsigned int neg_num;
  float w_seg[NCLASSES];
  float w_att[NCLASSES];
  float bce_wpos;            // weight for t==1 elements = neg/total
  float bce_wneg;            // weight for t==0 elements = pos/total
  float seg_num[NIMG];
  float seg_den[NIMG];
  float att_num[NIMG];
  float att_den[NIMG];
  float bce_sum;
};

// ---- 32-lane f32 allreduce via V_WMMA_F32_16X16X4_F32 -----------------
// A (16x4): a0 holds K=0 (lanes 0-15) / K=2 (lanes 16-31); a1 = 0.
// B = all-ones, so D[m][n] = rowsum_m = v[m] + v[m+16] (layout-independent).
// Stage 2: feed D vgpr pairs back as A columns, accumulating into C.
// Result: total sum of all 32 lane values, broadcast to every lane.
__device__ __forceinline__ float wave_sum_wmma(float v) {
  v2f a;   a[0] = v;    a[1] = 0.0f;
  v2f one; one[0] = 1.0f; one[1] = 1.0f;
  v8f z = {};
  v8f d = __builtin_amdgcn_wmma_f32_16x16x4_f32(false, a, false, one,
                                                (short)0, z, false, false);
  v8f acc = {};
  v2f a2;
  a2[0] = d[0]; a2[1] = d[1];
  acc = __builtin_amdgcn_wmma_f32_16x16x4_f32(false, a2, false, one,
                                              (short)0, acc, false, false);
  a2[0] = d[2]; a2[1] = d[3];
  acc = __builtin_amdgcn_wmma_f32_16x16x4_f32(false, a2, false, one,
                                              (short)0, acc, false, false);
  a2[0] = d[4]; a2[1] = d[5];
  acc = __builtin_amdgcn_wmma_f32_16x16x4_f32(false, a2, false, one,
                                              (short)0, acc, false, false);
  a2[0] = d[6]; a2[1] = d[7];
  acc = __builtin_amdgcn_wmma_f32_16x16x4_f32(false, a2, false, one,
                                              (short)0, acc, false, false);
  return acc[0];
}

// ---- kernel 1: histograms + BCE pos/neg counts ------------------------
__global__ __launch_bounds__(256) void k_hist(const int* __restrict__ segmask,
                                              const float* __restrict__ edgein,
                                              const int* __restrict__ edgemask,
                                              Ws* __restrict__ ws) {
  __shared__ unsigned int s_seg[NCLASSES];
  __shared__ unsigned int s_att[NCLASSES];
  __shared__ unsigned int s_pos, s_neg;
  const int tid = threadIdx.x;
  if (tid < NCLASSES) { s_seg[tid] = 0u; s_att[tid] = 0u; }
  if (tid == 0) { s_pos = 0u; s_neg = 0u; }
  __syncthreads();

  const int q = blockIdx.x * 256 + tid;
  int4   t4 = ((const int4*)segmask)[q];
  float4 e4 = ((const float4*)edgein)[q];
  int4   m4 = ((const int4*)edgemask)[q];

  int   ts[4] = {t4.x, t4.y, t4.z, t4.w};
  float es[4] = {e4.x, e4.y, e4.z, e4.w};
  int   ms[4] = {m4.x, m4.y, m4.z, m4.w};

  unsigned int pc = 0, nc = 0;
#pragma unroll
  for (int j = 0; j < 4; ++j) {
    int t = ts[j];
    bool inside = (t >= 0) && (t <= NCLASSES);   // torch.histc semantics
    int b = t < (NCLASSES - 1) ? t : (NCLASSES - 1);
    if (inside) {
      atomicAdd(&s_seg[b], 1u);
      if (es[j] > 0.8f) atomicAdd(&s_att[b], 1u);  // att_target!=IGNORE
    }
    pc += (ms[j] == 1) ? 1u : 0u;
    nc += (ms[j] == 0) ? 1u : 0u;
  }
  atomicAdd(&s_pos, pc);
  atomicAdd(&s_neg, nc);
  __syncthreads();

  if (tid < NCLASSES) {
    atomicAdd(&ws->seg_bins[tid], s_seg[tid]);
    atomicAdd(&ws->att_bins[tid], s_att[tid]);
  }
  if (tid == 0) {
    atomicAdd(&ws->pos_num, s_pos);
    atomicAdd(&ws->neg_num, s_neg);
  }
}

// ---- kernel 2: bins -> class weights + BCE coefficients ---------------
__global__ void k_weights(Ws* __restrict__ ws) {
  if (threadIdx.x == 0 && blockIdx.x == 0) {
    float ssum = 0.f, asum = 0.f;
    for (int c = 0; c < NCLASSES; ++c) {
      ssum += (float)ws->seg_bins[c];
      asum += (float)ws->att_bins[c];
    }
    for (int c = 0; c < NCLASSES; ++c) {
      float sb = (float)ws->seg_bins[c];
      float ab = (float)ws->att_bins[c];
      // (bins != 0) * UPPER_BOUND * (1 - bins/sum) + 1
      ws->w_seg[c] = (sb != 0.f ? (1.0f - sb / ssum) : 0.0f) + 1.0f;
      ws->w_att[c] = (ab != 0.f ? (1.0f - ab / asum) : 0.0f) + 1.0f;
    }
    float pos = (float)ws->pos_num;
    float neg = (float)ws->neg_num;
    float tot = pos + neg;
    ws->bce_wpos = neg / tot;
    ws->bce_wneg = pos / tot;
  }
}

// ---- kernel 3: streaming log-softmax + weighted NLL + BCE -------------
__global__ __launch_bounds__(256) void k_main(const float* __restrict__ segin,
                                              const float* __restrict__ edgein,
                                              const int* __restrict__ segmask,
                                              const int* __restrict__ edgemask,
                                              Ws* __restrict__ ws) {
  __shared__ float sh_wseg[NCLASSES];
  __shared__ float sh_watt[NCLASSES];
  __shared__ float sh_bcew[2];
  const int tid = threadIdx.x;
  if (tid < NCLASSES) {
    sh_wseg[tid] = ws->w_seg[tid];
    sh_watt[tid] = ws->w_att[tid];
  }
  if (tid == 0) { sh_bcew[0] = ws->bce_wpos; sh_bcew[1] = ws->bce_wneg; }
  __syncthreads();

  const int q = blockIdx.x * 256 + tid;       // quad-pixel index
  const int n = q >> 17;                       // sample (wave-uniform)
  const int hwq = q & (QHW - 1);

  const float4* segin4 = (const float4*)segin;
  const int base = n * NCLASSES * QHW + hwq;

  float4 v[NCLASSES];
#pragma unroll
  for (int c = 0; c < NCLASSES; ++c) v[c] = segin4[base + c * QHW];

  int4   t4 = ((const int4*)segmask)[q];
  float4 e4 = ((const float4*)edgein)[q];
  int4   m4 = ((const int4*)edgemask)[q];
  int   ts[4]  = {t4.x, t4.y, t4.z, t4.w};
  float es[4]  = {e4.x, e4.y, e4.z, e4.w};
  int   ems[4] = {m4.x, m4.y, m4.z, m4.w};

  // max over channels (4 pixels in parallel, all static indices)
  float mx = v[0].x, my = v[0].y, mz = v[0].z, mw = v[0].w;
#pragma unroll
  for (int c = 1; c < NCLASSES; ++c) {
    mx = fmaxf(mx, v[c].x); my = fmaxf(my, v[c].y);
    mz = fmaxf(mz, v[c].z); mw = fmaxf(mw, v[c].w);
  }
  // sum of exp + pick x[t] via select (no dynamic register indexing)
  float sx = 0.f, sy = 0.f, sz = 0.f, sw = 0.f;
  float px = 0.f, py = 0.f, pz = 0.f, pw = 0.f;
#pragma unroll
  for (int c = 0; c < NCLASSES; ++c) {
    sx += __expf(v[c].x - mx); sy += __expf(v[c].y - my);
    sz += __expf(v[c].z - mz); sw += __expf(v[c].w - mw);
    px = (c == ts[0]) ? v[c].x : px;
    py = (c == ts[1]) ? v[c].y : py;
    pz = (c == ts[2]) ? v[c].z : pz;
    pw = (c == ts[3]) ? v[c].w : pw;
  }
  float lsex = mx + __logf(sx);
  float lsey = my + __logf(sy);
  float lsez = mz + __logf(sz);
  float lsew = mw + __logf(sw);

  float segn = 0.f, segd = 0.f, attn = 0.f, attd = 0.f, bces = 0.f;
  auto docomp = [&](float lse, float pick, int t, float e, int em) {
    bool valid = (t != IGNORE_LBL) && (t >= 0) && (t < NCLASSES);
    int tc = valid ? t : 0;
    float nll = lse - pick;                       // -logp[t]
    float wv = sh_wseg[tc] * (valid ? 1.0f : 0.0f);
    segn += wv * nll;
    segd += wv;
    bool keep = e > 0.8f;
    float wa = sh_watt[tc] * ((valid && keep) ? 1.0f : 0.0f);
    attn += wa * nll;
    attd += wa;
    float x = e;
    float tt = (float)em;
    float bw = (em == 1) ? sh_bcew[0] : ((em == 0) ? sh_bcew[1] : 0.0f);
    float bce = fmaxf(x, 0.0f) - x * tt + __logf(1.0f + __expf(-fabsf(x)));
    bces += bw * bce;
  };
  docomp(lsex, px, ts[0], es[0], ems[0]);
  docomp(lsey, py, ts[1], es[1], ems[1]);
  docomp(lsez, pz, ts[2], es[2], ems[2]);
  docomp(lsew, pw, ts[3], es[3], ems[3]);

  // 5 wave allreduces through the matrix pipe (EXEC all-ones here)
  float r0 = wave_sum_wmma(segn);
  float r1 = wave_sum_wmma(segd);
  float r2 = wave_sum_wmma(attn);
  float r3 = wave_sum_wmma(attd);
  float r4 = wave_sum_wmma(bces);

  if ((tid & 31) == 0) {
    atomicAdd(&ws->seg_num[n], r0);
    atomicAdd(&ws->seg_den[n], r1);
    atomicAdd(&ws->att_num[n], r2);
    atomicAdd(&ws->att_den[n], r3);
    atomicAdd(&ws->bce_sum, r4);
  }
}

// ---- kernel 4: finalize 3 scalars -------------------------------------
__global__ void k_final(const Ws* __restrict__ ws, float* __restrict__ out) {
  if (threadIdx.x == 0 && blockIdx.x == 0) {
    float seg = 0.f, att = 0.f;
    for (int i = 0; i < NIMG; ++i) {
      seg += ws->seg_num[i] / ws->seg_den[i];
      att += ws->att_num[i] / ws->att_den[i];
    }
    float edge = 20.0f * ws->bce_sum / (float)PIX;
    out[0] = seg;
    out[1] = edge;
    out[2] = att;
  }
}

// ---- launcher ---------------------------------------------------------
extern "C" void kernel_launch(void* const* d_in, const int* in_sizes, int n_in,
                              void* d_out, int out_size, void* d_ws, size_t ws_size,
                              hipStream_t stream) {
  (void)in_sizes; (void)n_in; (void)out_size; (void)ws_size;
  const float* segin   = (const float*)d_in[0];
  const float* edgein  = (const float*)d_in[1];
  const int*   segmask = (const int*)d_in[2];
  const int*   edgemask= (const int*)d_in[3];
  Ws* ws = (Ws*)d_ws;
  float* out = (float*)d_out;

  hipMemsetAsync(d_ws, 0, sizeof(Ws), stream);

  dim3 blk(256);
  dim3 grd(QPIX / 256);   // 2048 blocks, exact cover, no grid-stride
  k_hist<<<grd, blk, 0, stream>>>(segmask, edgein, edgemask, ws);
  k_weights<<<1, 32, 0, stream>>>(ws);
  k_main<<<grd, blk, 0, stream>>>(segin, edgein, segmask, edgemask, ws);
  k_final<<<1, 32, 0, stream>>>(ws, out);
}